// EnergyScoreLoss_33809982554431
// MI455X (gfx1250) — compile-verified
//
#include <hip/hip_runtime.h>

typedef float v2f __attribute__((ext_vector_type(2)));
typedef float v8f __attribute__((ext_vector_type(8)));

#define NB 2048
#define MM 64
#define DD 512
#define KC 128               // K-chunk staged in LDS
#define NCHUNK (DD / KC)
#define SP (KC + 4)          // padded LDS row stride (floats): 132 % 64 = 4 -> conflict-free

__launch_bounds__(256)
__global__ void energy_main(const float* __restrict__ y,
                            const float* __restrict__ preds,
                            float* __restrict__ ws1,
                            float* __restrict__ ws2)
{
  __shared__ float Pc[MM * SP];   // 64*132*4 = 33.8 KB
  __shared__ float yv[DD];        // 2 KB
  __shared__ float sqv[MM];
  __shared__ float rowT1[MM];
  __shared__ float waveT2[8];

  const int n    = blockIdx.x;
  const int t    = threadIdx.x;
  const int lane = t & 31;        // wave32
  const int wave = t >> 5;

  const float* pn = preds + (size_t)n * (MM * DD);

  if (t < DD / 4)
    ((float4*)yv)[t] = ((const float4*)(y + (size_t)n * DD))[t];

  // phase-1 mapping: 4 threads per row, each owns a 32-wide K segment per chunk
  const int p1row = t >> 2;
  const int p1seg = t & 3;
  float sqp = 0.f, dyp = 0.f;

  // phase-2 mapping: wave w owns tiles 2w, 2w+1 of the 4x4 tile grid (same A row-block)
  const int it    = wave >> 1;
  const int jt0   = (wave & 1) * 2;
  const int rsel  = lane & 15;
  const int khalf = (lane >> 4) * 2;   // A/B 16x4 f32 layout: upper half-wave holds K+2,K+3
  const float* aP  = &Pc[(it * 16 + rsel) * SP + khalf];
  const float* bP0 = &Pc[(jt0 * 16 + rsel) * SP + khalf];
  const float* bP1 = &Pc[((jt0 + 1) * 16 + rsel) * SP + khalf];

  v8f c0 = {0.f,0.f,0.f,0.f,0.f,0.f,0.f,0.f};
  v8f c1 = {0.f,0.f,0.f,0.f,0.f,0.f,0.f,0.f};

  for (int kc = 0; kc < NCHUNK; ++kc) {
    __syncthreads();                       // previous chunk fully consumed
    // stage chunk into LDS: coalesced global_load_b128 -> ds_store_b128
    const float* src = pn + kc * KC;
    for (int g = t; g < MM * (KC / 4); g += 256) {
      const int row = g >> 5;              // 32 float4 per row-chunk
      const int c4  = (g & 31) << 2;
      const float4 v = *(const float4*)(src + row * DD + c4);
      *(float4*)&Pc[row * SP + c4] = v;
    }
    if (kc + 1 < NCHUNK && t < MM)         // warm next chunk: global_prefetch_b8
      __builtin_prefetch(src + KC + t * DD, 0, 0);
    __syncthreads();

    // ---- phase 1 partials: ||p||^2 and ||p - y||^2 (from LDS) ----
    {
      const float* pr = &Pc[p1row * SP + p1seg * 32];
      const float* yr = &yv[kc * KC + p1seg * 32];
      #pragma unroll
      for (int k = 0; k < 32; k += 4) {
        const float4 p = *(const float4*)(pr + k);
        const float4 q = *(const float4*)(yr + k);
        sqp += p.x*p.x + p.y*p.y + p.z*p.z + p.w*p.w;
        const float ax = p.x-q.x, ay = p.y-q.y, az = p.z-q.z, aw = p.w-q.w;
        dyp += ax*ax + ay*ay + az*az + aw*aw;
      }
    }

    // ---- phase 2 partials: Gram via V_WMMA_F32_16X16X4_F32 ----
    #pragma unroll 4
    for (int k = 0; k < KC; k += 4) {
      const v2f a  = *(const v2f*)(aP + k);
      const v2f b0 = *(const v2f*)(bP0 + k);
      const v2f b1 = *(const v2f*)(bP1 + k);
      c0 = __builtin_amdgcn_wmma_f32_16x16x4_f32(false, a, false, b0,
                                                 (short)0, c0, false, false);
      c1 = __builtin_amdgcn_wmma_f32_16x16x4_f32(false, a, false, b1,
                                                 (short)0, c1, false, false);
    }
  }

  // finalize phase 1: fold the 4 K-segments of each row (wave32 shuffles)
  sqp += __shfl_xor(sqp, 1, 32);
  sqp += __shfl_xor(sqp, 2, 32);
  dyp += __shfl_xor(dyp, 1, 32);
  dyp += __shfl_xor(dyp, 2, 32);
  if (p1seg == 0) {
    sqv[p1row]   = sqp;
    rowT1[p1row] = sqrtf(dyp);
  }
  __syncthreads();

  // distances from gram tiles: d2 = sq[i] + sq[j] - 2*gram ; mask diagonal
  float t2 = 0.f;
  {
    const int mofs = (lane >> 4) * 8;     // C layout: lanes 16-31 hold M = v+8
    const int col0 = jt0 * 16 + rsel;
    #pragma unroll
    for (int v = 0; v < 8; ++v) {
      const int r = it * 16 + mofs + v;
      const float d2a = sqv[r] + sqv[col0] - 2.f * c0[v];
      if (r != col0 && d2a > 0.f) t2 += sqrtf(d2a);
      const int cj = col0 + 16;
      const float d2b = sqv[r] + sqv[cj] - 2.f * c1[v];
      if (r != cj && d2b > 0.f) t2 += sqrtf(d2b);
    }
  }
  #pragma unroll
  for (int o = 16; o >= 1; o >>= 1) t2 += __shfl_xor(t2, o, 32);
  if (lane == 0) waveT2[wave] = t2;
  __syncthreads();

  if (t == 0) {
    float s1 = 0.f;
    #pragma unroll
    for (int i = 0; i < MM; ++i) s1 += rowT1[i];
    float s2 = 0.f;
    #pragma unroll
    for (int w = 0; w < 8; ++w) s2 += waveT2[w];
    ws1[n] = s1;                           // sum_m ||p - y||
    ws2[n] = s2;                           // sum_{i != j} ||p_i - p_j||
  }
}

__launch_bounds__(256)
__global__ void energy_finalize(const float* __restrict__ ws1,
                                const float* __restrict__ ws2,
                                float* __restrict__ out)
{
  __shared__ float s1s[256];
  __shared__ float s2s[256];
  const int t = threadIdx.x;
  float s1 = 0.f, s2 = 0.f;
  for (int i = t; i < NB; i += 256) {      // fixed order -> deterministic
    s1 += ws1[i];
    s2 += ws2[i];
  }
  s1s[t] = s1; s2s[t] = s2;
  __syncthreads();
  #pragma unroll
  for (int o = 128; o > 0; o >>= 1) {
    if (t < o) { s1s[t] += s1s[t + o]; s2s[t] += s2s[t + o]; }
    __syncthreads();
  }
  if (t == 0) {
    const double inv1 = 1.0 / ((double)NB * (double)MM);
    const double inv2 = 1.0 / ((double)NB * 2.0 * (double)MM * (double)(MM - 1) * (double)DD);
    out[0] = (float)((double)s1s[0] * inv1 - (double)s2s[0] * inv2);
  }
}

extern "C" void kernel_launch(void* const* d_in, const int* in_sizes, int n_in,
                              void* d_out, int out_size, void* d_ws, size_t ws_size,
                              hipStream_t stream) {
  (void)in_sizes; (void)n_in; (void)out_size; (void)ws_size;
  const float* y     = (const float*)d_in[0];
  const float* preds = (const float*)d_in[1];
  float* ws1 = (float*)d_ws;          // 2048 floats
  float* ws2 = ws1 + NB;              // 2048 floats
  energy_main<<<NB, 256, 0, stream>>>(y, preds, ws1, ws2);
  energy_finalize<<<1, 256, 0, stream>>>(ws1, ws2, (float*)d_out);
}